// unitary_diag_projection_26173530702291
// MI455X (gfx1250) — compile-verified
//
#include <hip/hip_runtime.h>

// Problem constants (match reference)
#define NBATCH 1024
#define INPUT  8
#define CHAN   4
#define HID    64
#define TERMS  18
#define MAXSQ  12
#define STRIDE 66   // padded LDS row stride (floats); even -> 8B-aligned pairs

typedef __attribute__((ext_vector_type(2))) float v2f;
typedef __attribute__((ext_vector_type(8))) float v8f;

__global__ __launch_bounds__(512)
void expm_unitary_kernel(const float* __restrict__ dX,   // (N, 8) f32
                         const float* __restrict__ A,    // (8, 4, 63) complex64 interleaved
                         float* __restrict__ out)        // (N, 4, 64, 64) complex64 interleaved
{
    __shared__ float sRe[2][HID][STRIDE];
    __shared__ float sIm[2][HID][STRIDE];
    __shared__ float aRe[HID], aIm[HID], absA[HID], red[HID];
    __shared__ float cR[HID], cI[HID];
    __shared__ float sScale;
    __shared__ int   sSq;

    const int tid = threadIdx.x;
    const int blk = blockIdx.x;          // blk = b*CHAN + c
    const int b   = blk >> 2;
    const int c   = blk & 3;

    // ---- phase 0: a_i = sum_m A[m,c,i] * dX[b,m]  (tridiagonal coefficients) ----
    if (tid < HID) {
        float ar = 0.f, ai = 0.f;
        if (tid < HID - 1) {
            #pragma unroll
            for (int m = 0; m < INPUT; ++m) {
                float x = dX[b * INPUT + m];
                int base = 2 * ((m * CHAN + c) * (HID - 1) + tid);
                ar = fmaf(A[base],     x, ar);
                ai = fmaf(A[base + 1], x, ai);
            }
        }
        aRe[tid] = ar; aIm[tid] = ai;
        absA[tid] = sqrtf(ar * ar + ai * ai);
    }
    __syncthreads();
    // column abs-sums of AX: colsum_j = 0.5*(|a_{j-1}| + |a_j|), a_{-1}=a_{63}=0
    if (tid < HID) {
        float left = (tid > 0) ? absA[tid - 1] : 0.f;
        red[tid] = 0.5f * (left + absA[tid]);
    }
    __syncthreads();
    if (tid == 0) {
        float nrm = 0.f;
        for (int j = 0; j < HID; ++j) nrm = fmaxf(nrm, red[j]);
        int s = 0;
        if (nrm > 1.0f) s = (int)ceilf(log2f(nrm));
        if (s < 0) s = 0;
        sScale = exp2f(-(float)s);
        sSq = (s > MAXSQ) ? MAXSQ : s;
    }
    __syncthreads();
    const float scl = sScale;
    const int   sq  = sSq;
    if (tid < HID) {            // c_i = 0.5*scale*a_i ; c_63 := 0
        cR[tid] = 0.5f * scl * aRe[tid];
        cI[tid] = 0.5f * scl * aIm[tid];
    }

    // ---- init E = I in buffer 0 ----
    {
        const int j = tid & 63;
        const int i0 = tid >> 6;
        #pragma unroll
        for (int t = 0; t < 8; ++t) {
            int i = i0 + 8 * t;
            sRe[0][i][j] = (i == j) ? 1.0f : 0.0f;
            sIm[0][i][j] = 0.0f;
        }
    }
    __syncthreads();

    // ---- Taylor (Horner): E = I + (T@E)/k ; T tridiagonal so T@E is O(n^2) ----
    // (T@E)[i,j] = -conj(c_{i-1})*E[i-1,j] + c_i*E[i+1,j]
    int cur = 0;
    for (int k = TERMS; k >= 1; --k) {
        const float kf = (float)k;
        const int nxt = cur ^ 1;
        const int j = tid & 63;
        const int i0 = tid >> 6;
        #pragma unroll
        for (int t = 0; t < 8; ++t) {
            int i   = i0 + 8 * t;
            int im1 = (i > 0)  ? i - 1 : 0;
            int ip1 = (i < 63) ? i + 1 : 63;
            float mu = (i > 0)  ? 1.f : 0.f;
            float mv = (i < 63) ? 1.f : 0.f;
            float ur = sRe[cur][im1][j] * mu, ui = sIm[cur][im1][j] * mu;
            float vr = sRe[cur][ip1][j] * mv, vi = sIm[cur][ip1][j] * mv;
            float c1r = cR[im1], c1i = cI[im1];
            float c2r = cR[i],   c2i = cI[i];
            float tr = (-c1r * ur - c1i * ui) + (c2r * vr - c2i * vi);
            float ti = (-c1r * ui + c1i * ur) + (c2r * vi + c2i * vr);
            sRe[nxt][i][j] = ((i == j) ? 1.0f : 0.0f) + tr / kf;
            sIm[nxt][i][j] = ti / kf;
        }
        __syncthreads();
        cur = nxt;
    }

    // ---- squarings: E <- E@E, s times, dense complex matmul on f32 WMMA ----
    // wave w owns output tile (w>>2, w&3); 3 accumulators: re = PP - MM, im = RI
    const int lane = tid & 31;
    const int wave = tid >> 5;
    const int half = lane >> 4;          // A/B fragment lane-half selects K pair
    const int lm   = lane & 15;
    const int trow = (wave >> 2) * 16;
    const int tcol = (wave & 3) * 16;
    const int arow = trow + lm;
    const int bcol = tcol + lm;

    for (int it = 0; it < sq; ++it) {    // sq is block-uniform -> EXEC all-ones
        const int nxt = cur ^ 1;
        v8f aPP = {}; v8f aMM = {}; v8f aRI = {};
        #pragma unroll
        for (int kk = 0; kk < 16; ++kk) {
            const int ka = kk * 4 + 2 * half;
            v2f Ar, Ai, Br, Bi;
            Ar.x = sRe[cur][arow][ka];     Ar.y = sRe[cur][arow][ka + 1];
            Ai.x = sIm[cur][arow][ka];     Ai.y = sIm[cur][arow][ka + 1];
            Br.x = sRe[cur][ka][bcol];     Br.y = sRe[cur][ka + 1][bcol];
            Bi.x = sIm[cur][ka][bcol];     Bi.y = sIm[cur][ka + 1][bcol];
            aPP = __builtin_amdgcn_wmma_f32_16x16x4_f32(false, Ar, false, Br, (short)0, aPP, false, false);
            aMM = __builtin_amdgcn_wmma_f32_16x16x4_f32(false, Ai, false, Bi, (short)0, aMM, false, false);
            aRI = __builtin_amdgcn_wmma_f32_16x16x4_f32(false, Ar, false, Bi, (short)0, aRI, false, false);
            aRI = __builtin_amdgcn_wmma_f32_16x16x4_f32(false, Ai, false, Br, (short)0, aRI, false, false);
        }
        // C/D layout: VGPR r -> row trow + r + 8*half, col tcol + lm
        #pragma unroll
        for (int r = 0; r < 8; ++r) {
            int row = trow + r + 8 * half;
            sRe[nxt][row][bcol] = aPP[r] - aMM[r];
            sIm[nxt][row][bcol] = aRI[r];
        }
        __syncthreads();
        cur = nxt;
    }

    // ---- write out: complex64 interleaved, (N,C,64,64) row-major ----
    {
        float2* o = (float2*)out;
        const size_t base = (size_t)blk * (HID * HID);
        const int j = tid & 63;
        const int i0 = tid >> 6;
        #pragma unroll
        for (int t = 0; t < 8; ++t) {
            int i = i0 + 8 * t;
            float2 v;
            v.x = sRe[cur][i][j];
            v.y = sIm[cur][i][j];
            o[base + (size_t)i * HID + j] = v;
        }
    }
}

extern "C" void kernel_launch(void* const* d_in, const int* in_sizes, int n_in,
                              void* d_out, int out_size, void* d_ws, size_t ws_size,
                              hipStream_t stream) {
    (void)in_sizes; (void)n_in; (void)out_size; (void)d_ws; (void)ws_size;
    const float* dX = (const float*)d_in[0];   // (1024, 8) float32
    const float* A  = (const float*)d_in[1];   // (8, 4, 63) complex64 -> interleaved floats
    float* out = (float*)d_out;                // (1024, 4, 64, 64) complex64 -> interleaved
    dim3 grid(NBATCH * CHAN);
    dim3 block(512);
    expm_unitary_kernel<<<grid, block, 0, stream>>>(dX, A, out);
}